// GNN_8942121910306
// MI455X (gfx1250) — compile-verified
//
#include <hip/hip_runtime.h>

#define N_NODES 100000
#define N_EDGES 600000
#define DIM     128
#define HID     256
#define LAYERS  5
#define EPS     1e-5f

typedef float v2f __attribute__((ext_vector_type(2)));
typedef float v8f __attribute__((ext_vector_type(8)));
typedef unsigned int u32x4 __attribute__((ext_vector_type(4)));
typedef int i32x4 __attribute__((ext_vector_type(4)));
typedef int i32x8 __attribute__((ext_vector_type(8)));

__device__ __forceinline__ float4 ld4(const float* p) {
    return *reinterpret_cast<const float4*>(p);
}

__device__ __forceinline__ void atomic_add_f32(float* p, float v) {
    __hip_atomic_fetch_add(p, v, __ATOMIC_RELAXED, __HIP_MEMORY_SCOPE_AGENT);
}

// ---------------------------------------------------------------------------
// TDM: stage a [32 x row_dwords] f32 tile (row-major, stride row_dwords) from
// global into LDS at `lds`, inserting pad_dwords=4 after every row so the LDS
// row stride becomes row_dwords+4 (bank-conflict-free WMMA fragment reads).
// D# layout per CDNA5 ISA sec 8.3/8.4.  pad_interval code: 6 -> 128 dwords,
// 7 -> 256 dwords.  Issued by one wave; TDM ignores EXEC.
// The low 32 bits of a flat shared-aperture pointer ARE the LDS byte offset
// (ISA 10.2 per-aperture calc), so they go straight into D#.lds_addr; the
// trailing empty asm both escapes the LDS array and clobbers memory so the
// compiler cannot fold the subsequent LDS fragment reads to undef.
// ---------------------------------------------------------------------------
__device__ __forceinline__ void tdm_load_tile_32rows(
    const float* gptr, float* lds, unsigned row_dwords, unsigned pad_code)
{
    unsigned long long ga = (unsigned long long)(size_t)gptr;
    unsigned lds_off = (unsigned)(size_t)lds;     // LDS byte offset (addr[31:0])
    u32x4 g0;
    g0[0] = 1u;                                   // count=1, user mode
    g0[1] = lds_off;                              // lds_addr
    g0[2] = (unsigned)(ga & 0xFFFFFFFFu);         // global_addr[31:0]
    g0[3] = (unsigned)(ga >> 32) | (2u << 30);    // global_addr[56:32] | type=2

    i32x8 g1;
    // data_size=4B(code2), pad_enable, pad_interval=pad_code, pad_amount=4dw(code3)
    g1[0] = (int)((2u << 16) | (1u << 20) | (pad_code << 22) | (3u << 25));
    g1[1] = (int)((row_dwords & 0xFFFFu) << 16);  // tensor_dim0[15:0]
    g1[2] = (int)((row_dwords >> 16) | (32u << 16)); // tensor_dim0[31:16] | tensor_dim1[15:0]=32
    g1[3] = (int)((row_dwords & 0xFFFFu) << 16);  // tensor_dim1[31:16]=0 | tile_dim0
    g1[4] = 32;                                   // tile_dim1=32 rows, tile_dim2=0
    g1[5] = (int)row_dwords;                      // tensor_dim0_stride[31:0]
    g1[6] = 0;                                    // stride hi | dim1_stride lo
    g1[7] = 0;

    i32x4 z4 = {0, 0, 0, 0};
#if __clang_major__ >= 23
    i32x8 z8 = {0, 0, 0, 0, 0, 0, 0, 0};
    __builtin_amdgcn_tensor_load_to_lds(g0, g1, z4, z4, z8, 0);
#else
    __builtin_amdgcn_tensor_load_to_lds(g0, g1, z4, z4, 0);
#endif
    __builtin_amdgcn_s_wait_tensorcnt(0);
    // Escape the LDS array + memory clobber: the TDM write is not visible to
    // alias analysis, so force the compiler to assume LDS was modified here.
    asm volatile("" : : "v"(lds) : "memory");
}

// ---------------------------------------------------------------------------
// Seed: h[n] = sum of six atom-feature embedding rows.  1 thread per float4.
// ---------------------------------------------------------------------------
__global__ __launch_bounds__(256) void k_seed(
    const int* __restrict__ an, const int* __restrict__ fc,
    const int* __restrict__ ct, const int* __restrict__ hy,
    const int* __restrict__ nh, const int* __restrict__ ar,
    const float* __restrict__ e_an, const float* __restrict__ e_fc,
    const float* __restrict__ e_ct, const float* __restrict__ e_hy,
    const float* __restrict__ e_nh, const float* __restrict__ e_ar,
    float* __restrict__ h)
{
    int idx = blockIdx.x * 256 + threadIdx.x;
    if (idx >= N_NODES * 32) return;
    int n   = idx >> 5;
    int col = (idx & 31) * 4;
    float4 a = ld4(e_an + (size_t)an[n] * DIM + col);
    float4 b = ld4(e_fc + (size_t)fc[n] * DIM + col);
    float4 c = ld4(e_ct + (size_t)ct[n] * DIM + col);
    float4 d = ld4(e_hy + (size_t)hy[n] * DIM + col);
    float4 e = ld4(e_nh + (size_t)nh[n] * DIM + col);
    float4 f = ld4(e_ar + (size_t)ar[n] * DIM + col);
    float4 o;
    o.x = a.x + b.x + c.x + d.x + e.x + f.x;
    o.y = a.y + b.y + c.y + d.y + e.y + f.y;
    o.z = a.z + b.z + c.z + d.z + e.z + f.z;
    o.w = a.w + b.w + c.w + d.w + e.w + f.w;
    *reinterpret_cast<float4*>(h + (size_t)n * DIM + col) = o;
}

// ---------------------------------------------------------------------------
// agg[n] = h[n] + self_vec   (self-loop uses bond-feature index 0)
// ---------------------------------------------------------------------------
__global__ __launch_bounds__(256) void k_agg_init(
    const float* __restrict__ h,
    const float* __restrict__ ec, const float* __restrict__ ea,
    const float* __restrict__ eb, const float* __restrict__ ed,
    const float* __restrict__ es,
    float* __restrict__ agg)
{
    int idx = blockIdx.x * 256 + threadIdx.x;
    if (idx >= N_NODES * 32) return;
    int n   = idx >> 5;
    int col = (idx & 31) * 4;
    float4 hv = ld4(h + (size_t)n * DIM + col);
    float4 a = ld4(ec + col);
    float4 b = ld4(ea + col);
    float4 c = ld4(eb + col);
    float4 d = ld4(ed + col);
    float4 e = ld4(es + col);
    float4 o;
    o.x = hv.x + a.x + b.x + c.x + d.x + e.x;
    o.y = hv.y + a.y + b.y + c.y + d.y + e.y;
    o.z = hv.z + a.z + b.z + c.z + d.z + e.z;
    o.w = hv.w + a.w + b.w + c.w + d.w + e.w;
    *reinterpret_cast<float4*>(agg + (size_t)n * DIM + col) = o;
}

// ---------------------------------------------------------------------------
// Edge messages + scatter-add.  One wave per edge (lane = float4 chunk):
// wave-uniform metadata loads, 512B coalesced row reads, f32 HW atomics.
// h (51MB) and agg (51MB) are L2-resident (192MB L2), so the random gather
// and the atomics resolve in L2, not HBM.
// ---------------------------------------------------------------------------
__global__ __launch_bounds__(256) void k_scatter(
    const float* __restrict__ h,
    const int* __restrict__ edge_index,
    const int* __restrict__ ic, const int* __restrict__ ia,
    const int* __restrict__ bt, const int* __restrict__ bd,
    const int* __restrict__ bs,
    const float* __restrict__ ec, const float* __restrict__ ea,
    const float* __restrict__ eb, const float* __restrict__ ed,
    const float* __restrict__ es,
    float* __restrict__ agg)
{
    int idx = blockIdx.x * 256 + threadIdx.x;
    if (idx >= N_EDGES * 32) return;
    int e   = idx >> 5;
    int col = (idx & 31) * 4;
    int src = edge_index[e];
    int dst = edge_index[N_EDGES + e];
    float4 hv = ld4(h + (size_t)src * DIM + col);
    float4 a = ld4(ec + (size_t)ic[e] * DIM + col);
    float4 b = ld4(ea + (size_t)ia[e] * DIM + col);
    float4 c = ld4(eb + (size_t)bt[e] * DIM + col);
    float4 d = ld4(ed + (size_t)bd[e] * DIM + col);
    float4 f = ld4(es + (size_t)bs[e] * DIM + col);
    float m0 = hv.x + a.x + b.x + c.x + d.x + f.x;
    float m1 = hv.y + a.y + b.y + c.y + d.y + f.y;
    float m2 = hv.z + a.z + b.z + c.z + d.z + f.z;
    float m3 = hv.w + a.w + b.w + c.w + d.w + f.w;
    float* p = agg + (size_t)dst * DIM + col;
    atomic_add_f32(p + 0, m0);
    atomic_add_f32(p + 1, m1);
    atomic_add_f32(p + 2, m2);
    atomic_add_f32(p + 3, m3);
}

// ---------------------------------------------------------------------------
// Pack W[K,OUT] into row-pair-interleaved Wp so a WMMA B-fragment (two
// consecutive K rows at one column) is a single contiguous b64 load:
//   Wp[(k2*OUT + c)*2 + {0,1}] = W[(2*k2 + {0,1})*OUT + c]
// ---------------------------------------------------------------------------
__global__ __launch_bounds__(256) void k_pack_w(
    const float* __restrict__ W, float* __restrict__ Wp, int K, int OUT)
{
    int idx = blockIdx.x * 256 + threadIdx.x;
    int tot = (K >> 1) * OUT;
    if (idx >= tot) return;
    int k2 = idx / OUT;
    int c  = idx - k2 * OUT;
    float2 v;
    v.x = W[(size_t)(2 * k2) * OUT + c];
    v.y = W[(size_t)(2 * k2 + 1) * OUT + c];
    *reinterpret_cast<float2*>(Wp + (size_t)idx * 2) = v;
}

// ---------------------------------------------------------------------------
// GEMM1: hid[N,256] = relu(agg[N,128] @ W1 + b1)  via V_WMMA_F32_16X16X4_F32.
// Block tile = 32 rows x 128 cols (grid.y selects col half); 8 waves, each
// owns one 16-col slice and BOTH 16-row tiles -> every B fragment feeds two
// WMMAs (ILP-2, half the B traffic).  A tile staged by the Tensor Data Mover
// with pad_interval/pad_amount giving LDS stride 132 dwords.
// ---------------------------------------------------------------------------
#define LDA1 132
__global__ __launch_bounds__(256) void k_gemm1(
    const float* __restrict__ A, const float* __restrict__ Wp,
    const float* __restrict__ bias, float* __restrict__ out)
{
    __shared__ float As[32 * LDA1];
    const int row0 = blockIdx.x * 32;
    if (threadIdx.x < 32) {
        tdm_load_tile_32rows(A + (size_t)row0 * DIM, As, DIM, 6u);
    }
    __syncthreads();

    const int wave = threadIdx.x >> 5;
    const int lane = threadIdx.x & 31;
    const int hi   = lane >> 4;                   // half-wave (K-pair select)
    const int mn   = lane & 15;                   // M (A) / N (B) index
    const int col  = blockIdx.y * 128 + wave * 16 + mn;

    v8f acc0 = {};
    v8f acc1 = {};
    for (int k = 0; k < DIM; k += 4) {
        v2f a0 = *reinterpret_cast<const v2f*>(&As[mn * LDA1 + k + 2 * hi]);
        v2f a1 = *reinterpret_cast<const v2f*>(&As[(16 + mn) * LDA1 + k + 2 * hi]);
        v2f b  = *reinterpret_cast<const v2f*>(
                     Wp + ((size_t)((k >> 1) + hi) * HID + col) * 2);
        acc0 = __builtin_amdgcn_wmma_f32_16x16x4_f32(false, a0, false, b,
                                                     (short)0, acc0, false, false);
        acc1 = __builtin_amdgcn_wmma_f32_16x16x4_f32(false, a1, false, b,
                                                     (short)0, acc1, false, false);
    }
    float bv = bias[col];
#pragma unroll
    for (int j = 0; j < 8; ++j) {
        float v0 = acc0[j] + bv;
        float v1 = acc1[j] + bv;
        v0 = v0 > 0.f ? v0 : 0.f;
        v1 = v1 > 0.f ? v1 : 0.f;
        out[(size_t)(row0 + j + 8 * hi) * HID + col]      = v0;
        out[(size_t)(row0 + 16 + j + 8 * hi) * HID + col] = v1;
    }
}

// ---------------------------------------------------------------------------
// GEMM2: h2[N,128] = hid[N,256] @ W2 + b2.  Block tile = 32 rows x 128 cols.
// LDS stride 260 dwords via TDM pad (interval code 7 = 256 dwords).
// ---------------------------------------------------------------------------
#define LDA2 260
__global__ __launch_bounds__(256) void k_gemm2(
    const float* __restrict__ A, const float* __restrict__ Wp,
    const float* __restrict__ bias, float* __restrict__ out)
{
    __shared__ float As[32 * LDA2];
    const int row0 = blockIdx.x * 32;
    if (threadIdx.x < 32) {
        tdm_load_tile_32rows(A + (size_t)row0 * HID, As, HID, 7u);
    }
    __syncthreads();

    const int wave = threadIdx.x >> 5;
    const int lane = threadIdx.x & 31;
    const int hi   = lane >> 4;
    const int mn   = lane & 15;
    const int col  = wave * 16 + mn;

    v8f acc0 = {};
    v8f acc1 = {};
    for (int k = 0; k < HID; k += 4) {
        v2f a0 = *reinterpret_cast<const v2f*>(&As[mn * LDA2 + k + 2 * hi]);
        v2f a1 = *reinterpret_cast<const v2f*>(&As[(16 + mn) * LDA2 + k + 2 * hi]);
        v2f b  = *reinterpret_cast<const v2f*>(
                     Wp + ((size_t)((k >> 1) + hi) * DIM + col) * 2);
        acc0 = __builtin_amdgcn_wmma_f32_16x16x4_f32(false, a0, false, b,
                                                     (short)0, acc0, false, false);
        acc1 = __builtin_amdgcn_wmma_f32_16x16x4_f32(false, a1, false, b,
                                                     (short)0, acc1, false, false);
    }
    float bv = bias[col];
#pragma unroll
    for (int j = 0; j < 8; ++j) {
        out[(size_t)(row0 + j + 8 * hi) * DIM + col]      = acc0[j] + bv;
        out[(size_t)(row0 + 16 + j + 8 * hi) * DIM + col] = acc1[j] + bv;
    }
}

// ---------------------------------------------------------------------------
// BatchNorm: stats (sum, sumsq per channel) then normalize (+optional relu)
// ---------------------------------------------------------------------------
__global__ void k_zero_stats(float* __restrict__ stats)
{
    stats[threadIdx.x] = 0.f;
}

#define RPB 512
__global__ __launch_bounds__(128) void k_bn_reduce(
    const float* __restrict__ X, float* __restrict__ stats)
{
    int c  = threadIdx.x;                 // channel
    int r0 = blockIdx.x * RPB;
    float s = 0.f, q = 0.f;
    for (int r = 0; r < RPB; ++r) {
        int rr = r0 + r;
        if (rr < N_NODES) {
            float v = X[(size_t)rr * DIM + c];
            s += v;
            q += v * v;
        }
    }
    atomic_add_f32(&stats[c], s);
    atomic_add_f32(&stats[DIM + c], q);
}

__global__ __launch_bounds__(256) void k_bn_apply(
    const float* __restrict__ X, const float* __restrict__ stats,
    const float* __restrict__ gamma, const float* __restrict__ beta,
    float* __restrict__ out, int do_relu)
{
    int idx = blockIdx.x * 256 + threadIdx.x;
    if (idx >= N_NODES * 32) return;
    int n   = idx >> 5;
    int col = (idx & 31) * 4;
    const float inv_n = 1.0f / (float)N_NODES;
    float4 v = ld4(X + (size_t)n * DIM + col);
    float o[4] = {v.x, v.y, v.z, v.w};
#pragma unroll
    for (int j = 0; j < 4; ++j) {
        int c = col + j;
        float mean = stats[c] * inv_n;
        float var  = stats[DIM + c] * inv_n - mean * mean;
        float r    = __frsqrt_rn(var + EPS);
        float y    = gamma[c] * (o[j] - mean) * r + beta[c];
        if (do_relu) y = y > 0.f ? y : 0.f;
        o[j] = y;
    }
    float4 ov = {o[0], o[1], o[2], o[3]};
    *reinterpret_cast<float4*>(out + (size_t)n * DIM + col) = ov;
}

// ---------------------------------------------------------------------------
extern "C" void kernel_launch(void* const* d_in, const int* in_sizes, int n_in,
                              void* d_out, int out_size, void* d_ws, size_t ws_size,
                              hipStream_t stream)
{
    const int* atomic_num   = (const int*)d_in[0];
    const int* formal_chg   = (const int*)d_in[1];
    const int* chiral_tag   = (const int*)d_in[2];
    const int* hybrid       = (const int*)d_in[3];
    const int* num_hs       = (const int*)d_in[4];
    const int* is_arom      = (const int*)d_in[5];
    const int* edge_index   = (const int*)d_in[6];
    const int* is_conj      = (const int*)d_in[7];
    const int* e_is_arom    = (const int*)d_in[8];
    const int* bond_type    = (const int*)d_in[9];
    const int* bond_dir     = (const int*)d_in[10];
    const int* bond_stereo  = (const int*)d_in[11];
    const float* emb_an     = (const float*)d_in[12];
    const float* emb_fc     = (const float*)d_in[13];
    const float* emb_ct     = (const float*)d_in[14];
    const float* emb_hy     = (const float*)d_in[15];
    const float* emb_nh     = (const float*)d_in[16];
    const float* emb_ar     = (const float*)d_in[17];
    const float* e_conj     = (const float*)d_in[18];
    const float* e_arom     = (const float*)d_in[19];
    const float* e_btype    = (const float*)d_in[20];
    const float* e_bdir     = (const float*)d_in[21];
    const float* e_bstereo  = (const float*)d_in[22];
    const float* mlp_w1     = (const float*)d_in[23];
    const float* mlp_b1     = (const float*)d_in[24];
    const float* mlp_w2     = (const float*)d_in[25];
    const float* mlp_b2     = (const float*)d_in[26];
    const float* bn_gamma   = (const float*)d_in[27];
    const float* bn_beta    = (const float*)d_in[28];

    float* h     = (float*)d_out;                          // [N,128] persistent
    float* ws    = (float*)d_ws;
    float* agg   = ws;                                     // [N,128] (also GEMM2 out)
    float* hid   = agg + (size_t)N_NODES * DIM;            // [N,256]
    float* stats = hid + (size_t)N_NODES * HID;            // [256]
    float* w1p   = stats + 256;                            // [64*256*2]
    float* w2p   = w1p + (size_t)(DIM / 2) * HID * 2;      // [128*128*2]

    const int node_chunks = N_NODES * 32;
    const int edge_chunks = N_EDGES * 32;

    k_seed<<<(node_chunks + 255) / 256, 256, 0, stream>>>(
        atomic_num, formal_chg, chiral_tag, hybrid, num_hs, is_arom,
        emb_an, emb_fc, emb_ct, emb_hy, emb_nh, emb_ar, h);

    for (int l = 0; l < LAYERS; ++l) {
        const float* ec = e_conj    + (size_t)l * 3 * DIM;
        const float* ea = e_arom    + (size_t)l * 3 * DIM;
        const float* eb = e_btype   + (size_t)l * 23 * DIM;
        const float* ed = e_bdir    + (size_t)l * 8 * DIM;
        const float* es = e_bstereo + (size_t)l * 7 * DIM;
        const float* w1 = mlp_w1 + (size_t)l * DIM * HID;
        const float* b1 = mlp_b1 + (size_t)l * HID;
        const float* w2 = mlp_w2 + (size_t)l * HID * DIM;
        const float* b2 = mlp_b2 + (size_t)l * DIM;

        k_pack_w<<<((DIM / 2) * HID + 255) / 256, 256, 0, stream>>>(w1, w1p, DIM, HID);
        k_pack_w<<<((HID / 2) * DIM + 255) / 256, 256, 0, stream>>>(w2, w2p, HID, DIM);

        k_agg_init<<<(node_chunks + 255) / 256, 256, 0, stream>>>(
            h, ec, ea, eb, ed, es, agg);

        k_scatter<<<(edge_chunks + 255) / 256, 256, 0, stream>>>(
            h, edge_index, is_conj, e_is_arom, bond_type, bond_dir, bond_stereo,
            ec, ea, eb, ed, es, agg);

        k_gemm1<<<dim3(N_NODES / 32, 2), 256, 0, stream>>>(agg, w1p, b1, hid);

        k_zero_stats<<<1, 256, 0, stream>>>(stats);

        k_gemm2<<<dim3(N_NODES / 32), 256, 0, stream>>>(hid, w2p, b2, agg);

        k_bn_reduce<<<(N_NODES + RPB - 1) / RPB, 128, 0, stream>>>(agg, stats);

        k_bn_apply<<<(node_chunks + 255) / 256, 256, 0, stream>>>(
            agg, stats, bn_gamma + (size_t)l * DIM, bn_beta + (size_t)l * DIM,
            h, (l < LAYERS - 1) ? 1 : 0);
    }
}